// StableStateSpaceScan_40931038331136
// MI455X (gfx1250) — compile-verified
//
#include <hip/hip_runtime.h>
#include <cstdint>
#include <cstddef>

// ---------------- problem constants (from reference) ----------------
#define B_      8
#define K_      4096
#define C_      1024
#define CLAMP_V 1.0e4f

// ---------------- decomposition constants ----------------
#define NCHUNK  32
#define LCH     (K_ / NCHUNK)        // 128 steps per chunk (== 2^7)
#define TROWS   8                    // time-rows per async tile
#define NTILES  (LCH / TROWS)        // 16 tiles per chunk
#define BLOCK   256
#define WAVESZ  32
#define WAVES   (BLOCK / WAVESZ)     // 8 waves / block
#define VECC    4                    // channels per thread
#define WAVE_CH (WAVESZ * VECC)      // 128 channels per wave
#define ROWB    (C_ * 4)             // 4096 B per (b,k) row
#define WROWB   (WAVE_CH * 4)        // 512 B per wave per row
#define TILEB   (TROWS * WROWB)      // 4096 B per wave per buffer
#define LDSB    (WAVES * 2 * TILEB)  // 65536 B dynamic LDS per block

typedef float v4f __attribute__((ext_vector_type(4)));

// ---------------- helpers ----------------
__device__ __forceinline__ float sanitize1(float x) {
    x = (x != x) ? 0.0f : x;                 // NaN -> 0
    x = fmaxf(x, -CLAMP_V);                  // -inf -> -1e4
    x = fminf(x,  CLAMP_V);                  // +inf -> +1e4
    return x;
}
__device__ __forceinline__ float sigmoid1(float x) { return 1.0f / (1.0f + expf(-x)); }
__device__ __forceinline__ float softplus1(float x) { return (x > 20.0f) ? x : log1pf(expf(x)); }
__device__ __forceinline__ float decay1(float logit) {
    return 0.001f + 0.998f * sigmoid1(logit);   // MIN + (MAX-MIN)*sigmoid
}

// CDNA5 async global->LDS copy: each lane DMAs 16B; tracked by ASYNCcnt.
__device__ __forceinline__ void issue_tile_async(uint32_t lds_lane, uint64_t gbase,
                                                 uint32_t voff_lane) {
#pragma unroll
    for (int r = 0; r < TROWS; ++r) {
        asm volatile("global_load_async_to_lds_b128 %0, %1, %2"
                     :
                     : "v"(lds_lane + (uint32_t)(r * WROWB)),
                       "v"(voff_lane + (uint32_t)(r * ROWB)),
                       "s"(gbase)
                     : "memory");
    }
}
#define WAIT_ASYNC(N) asm volatile("s_wait_asynccnt %0" ::"n"(N) : "memory")

// ---------------- pass 1: per-chunk partial scan from zero state ----------------
__global__ __launch_bounds__(BLOCK)
void ssm_pass1(const float* __restrict__ tok, const float* __restrict__ dlog,
               const float* __restrict__ igr, const float* __restrict__ sbias,
               float* __restrict__ part) {
    extern __shared__ v4f lds[];
    const int bid   = blockIdx.x;
    const int b     = bid / NCHUNK;
    const int chunk = bid % NCHUNK;
    const int t     = threadIdx.x;
    const int wave  = t >> 5;
    const int lane  = t & 31;
    const int c0    = t * VECC;
    const int k0    = chunk * LCH;

    // per-channel constrained params (2 FMA form: bb = gp*x + bp ; s = a*s + bb)
    v4f dl = *(const v4f*)(dlog + c0);
    v4f ig = *(const v4f*)(igr + c0);
    v4f bs = *(const v4f*)(sbias + c0);
    float a[4], gp[4], bp[4];
#pragma unroll
    for (int i = 0; i < 4; ++i) {
        float d   = decay1(dl[i]);
        float g   = softplus1(ig[i]) + 1e-6f;
        float omd = 1.0f - d;
        a[i] = d; gp[i] = omd * g; bp[i] = omd * bs[i];
    }

    const uint32_t lds0   = (uint32_t)(uintptr_t)(&lds[0]);
    const uint32_t ldsA   = lds0 + (uint32_t)((wave * 2 + 0) * TILEB) + (uint32_t)(lane * 16);
    const uint32_t ldsB   = lds0 + (uint32_t)((wave * 2 + 1) * TILEB) + (uint32_t)(lane * 16);
    const uint32_t choff  = (uint32_t)(wave * WROWB + lane * 16);
    const uint64_t gbase  = (uint64_t)(uintptr_t)tok +
                            (uint64_t)((size_t)(b * K_ + k0) * C_) * 4ull;

    float s0 = 0.f, s1 = 0.f, s2 = 0.f, s3 = 0.f;

    issue_tile_async(ldsA, gbase, choff);                 // prime tile 0
    for (int tile = 0; tile < NTILES; ++tile) {
        const int cur = tile & 1;
        if (tile + 1 < NTILES) {
            issue_tile_async(cur ? ldsA : ldsB, gbase,
                             choff + (uint32_t)((tile + 1) * TROWS) * ROWB);
            WAIT_ASYNC(TROWS);                            // previous batch landed
        } else {
            WAIT_ASYNC(0);
        }
        const v4f* buf = lds + (size_t)(wave * 2 + cur) * (TROWS * 32) + lane;
#pragma unroll
        for (int r = 0; r < TROWS; ++r) {
            v4f x = buf[r * 32];
            s0 = fmaf(a[0], s0, fmaf(gp[0], sanitize1(x.x), bp[0]));
            s1 = fmaf(a[1], s1, fmaf(gp[1], sanitize1(x.y), bp[1]));
            s2 = fmaf(a[2], s2, fmaf(gp[2], sanitize1(x.z), bp[2]));
            s3 = fmaf(a[3], s3, fmaf(gp[3], sanitize1(x.w), bp[3]));
        }
    }

    v4f p; p.x = s0; p.y = s1; p.z = s2; p.w = s3;
    *(v4f*)(part + (size_t)(b * NCHUNK + chunk) * C_ + c0) = p;
}

// ---------------- combine: fold chunk partials into chunk-start states ----------------
__global__ __launch_bounds__(BLOCK)
void ssm_combine(const float* __restrict__ state, const float* __restrict__ dlog,
                 const float* __restrict__ part, float* __restrict__ sstart,
                 float* __restrict__ fstate) {
    const int idx = blockIdx.x * BLOCK + threadIdx.x;   // [0, B*C)
    const int b = idx / C_, c = idx % C_;
    float d  = decay1(dlog[c]);
    float dL = d;
#pragma unroll
    for (int i = 0; i < 7; ++i) dL *= dL;               // d^(2^7) = d^LCH
    float s = state[idx];
    for (int j = 0; j < NCHUNK; ++j) {
        const size_t o = (size_t)(b * NCHUNK + j) * C_ + c;
        sstart[o] = s;
        s = fmaf(dL, s, part[o]);
    }
    fstate[idx] = s;                                    // raw final state (no gain)
}

// ---------------- pass 2: rescan chunks from true start states, emit outputs ----------------
__global__ __launch_bounds__(BLOCK)
void ssm_pass2(const float* __restrict__ tok, const float* __restrict__ dlog,
               const float* __restrict__ igr, const float* __restrict__ ogr,
               const float* __restrict__ sbias, const float* __restrict__ sstart,
               float* __restrict__ outp) {
    extern __shared__ v4f lds[];
    const int bid   = blockIdx.x;
    const int b     = bid / NCHUNK;
    const int chunk = bid % NCHUNK;
    const int t     = threadIdx.x;
    const int wave  = t >> 5;
    const int lane  = t & 31;
    const int c0    = t * VECC;
    const int k0    = chunk * LCH;

    v4f dl = *(const v4f*)(dlog + c0);
    v4f ig = *(const v4f*)(igr + c0);
    v4f og4 = *(const v4f*)(ogr + c0);
    v4f bs = *(const v4f*)(sbias + c0);
    float a[4], gp[4], bp[4], og[4];
#pragma unroll
    for (int i = 0; i < 4; ++i) {
        float d   = decay1(dl[i]);
        float g   = softplus1(ig[i]) + 1e-6f;
        float omd = 1.0f - d;
        a[i] = d; gp[i] = omd * g; bp[i] = omd * bs[i];
        og[i] = softplus1(og4[i]) + 1e-6f;
    }

    v4f sv = *(const v4f*)(sstart + (size_t)(b * NCHUNK + chunk) * C_ + c0);
    float s0 = sv.x, s1 = sv.y, s2 = sv.z, s3 = sv.w;

    const uint32_t lds0  = (uint32_t)(uintptr_t)(&lds[0]);
    const uint32_t ldsA  = lds0 + (uint32_t)((wave * 2 + 0) * TILEB) + (uint32_t)(lane * 16);
    const uint32_t ldsB  = lds0 + (uint32_t)((wave * 2 + 1) * TILEB) + (uint32_t)(lane * 16);
    const uint32_t choff = (uint32_t)(wave * WROWB + lane * 16);
    const uint64_t gbase = (uint64_t)(uintptr_t)tok +
                           (uint64_t)((size_t)(b * K_ + k0) * C_) * 4ull;

    issue_tile_async(ldsA, gbase, choff);
    for (int tile = 0; tile < NTILES; ++tile) {
        const int cur = tile & 1;
        if (tile + 1 < NTILES) {
            issue_tile_async(cur ? ldsA : ldsB, gbase,
                             choff + (uint32_t)((tile + 1) * TROWS) * ROWB);
            WAIT_ASYNC(TROWS);
        } else {
            WAIT_ASYNC(0);
        }
        const v4f* buf = lds + (size_t)(wave * 2 + cur) * (TROWS * 32) + lane;
#pragma unroll
        for (int r = 0; r < TROWS; ++r) {
            v4f x = buf[r * 32];
            s0 = fmaf(a[0], s0, fmaf(gp[0], sanitize1(x.x), bp[0]));
            s1 = fmaf(a[1], s1, fmaf(gp[1], sanitize1(x.y), bp[1]));
            s2 = fmaf(a[2], s2, fmaf(gp[2], sanitize1(x.z), bp[2]));
            s3 = fmaf(a[3], s3, fmaf(gp[3], sanitize1(x.w), bp[3]));
            v4f o; o.x = og[0] * s0; o.y = og[1] * s1; o.z = og[2] * s2; o.w = og[3] * s3;
            const int kk = k0 + tile * TROWS + r;
            // NT store: don't evict the L2-resident token stream with outputs
            __builtin_nontemporal_store(
                o, (v4f*)(outp + (size_t)(b * K_ + kk) * C_ + c0));
        }
    }
}

// ---------------- host-side launch ----------------
extern "C" void kernel_launch(void* const* d_in, const int* in_sizes, int n_in,
                              void* d_out, int out_size, void* d_ws, size_t ws_size,
                              hipStream_t stream) {
    (void)in_sizes; (void)n_in; (void)out_size; (void)ws_size;
    const float* tok  = (const float*)d_in[0];  // [B,K,C]
    const float* st   = (const float*)d_in[1];  // [B,C]
    const float* dl   = (const float*)d_in[2];  // [C]
    const float* igr  = (const float*)d_in[3];  // [C]
    const float* ogr  = (const float*)d_in[4];  // [C]
    const float* sb   = (const float*)d_in[5];  // [C]

    float* outp   = (float*)d_out;                         // [B,K,C]
    float* fstate = outp + (size_t)B_ * K_ * C_;           // [B,C]

    float* part   = (float*)d_ws;                          // [B,NCHUNK,C]  (1 MB)
    float* sstart = part + (size_t)B_ * NCHUNK * C_;       // [B,NCHUNK,C]  (1 MB)

    ssm_pass1<<<dim3(B_ * NCHUNK), dim3(BLOCK), LDSB, stream>>>(tok, dl, igr, sb, part);
    ssm_combine<<<dim3((B_ * C_) / BLOCK), dim3(BLOCK), 0, stream>>>(st, dl, part, sstart, fstate);
    ssm_pass2<<<dim3(B_ * NCHUNK), dim3(BLOCK), LDSB, stream>>>(tok, dl, igr, ogr, sb, sstart, outp);
}